// GAT_77945066488463
// MI455X (gfx1250) — compile-verified
//
#include <hip/hip_runtime.h>
#include <hip/hip_bf16.h>

// ---------------------------------------------------------------------------
// GAT (2-layer, 8-head) for gfx1250 using V_WMMA_F32_16X16X4_F32.
// Bandwidth-bound on the [8,2048,2048] attention tensors; fp32 WMMA keeps the
// GEMM reductions exact while still using the CDNA5 matrix pipeline.
// B operands staged in LDS per 64-K tile (block-shared, coalesced b128 loads).
// ---------------------------------------------------------------------------

#define N_NODES 2048
#define NFEAT   512
#define NHID    64
#define NHEADS  8
#define LRELU_A 0.2f
#define NEG_INF (-9.0e15f)

#define KT      64    // K-tile rows staged in LDS
#define BSTRIDE 72    // padded LDS row stride (floats): 16B aligned rows,
                      // kp=2 half-wave offset = 144 floats = +16 banks -> no conflicts

typedef __attribute__((ext_vector_type(2))) float v2f;
typedef __attribute__((ext_vector_type(8))) float v8f;

__device__ __forceinline__ v8f wmma_f32x4(v2f a, v2f b, v8f c) {
  // D = A(16x4 f32) * B(4x16 f32) + C(16x16 f32)
  return __builtin_amdgcn_wmma_f32_16x16x4_f32(
      /*neg_a=*/false, a, /*neg_b=*/false, b,
      /*c_mod=*/(short)0, c, /*reuse_a=*/false, /*reuse_b=*/false);
}

__device__ __forceinline__ float elu1(float x) {
  return x > 0.0f ? x : (expf(x) - 1.0f);
}
__device__ __forceinline__ float elu2(float x) {
  // reference applies elu inside the bank AND once more outside
  return elu1(elu1(x));
}

// Stage a KT x 64 f32 B-tile (contiguous rows, row stride 64 in global) into
// LDS with padded row stride BSTRIDE. Block-wide, coalesced 128-bit transfers.
__device__ __forceinline__ void stage_b_tile(const float* __restrict__ src,
                                             float* Bs /* __shared__ */) {
#pragma unroll
  for (int p = 0; p < (KT * NHID) / (256 * 4); ++p) {     // 4 passes
    const int off = p * 1024 + threadIdx.x * 4;           // tile-flat offset
    const int row = off >> 6;
    const int col = off & 63;
    const float4 v = *(const float4*)(src + off);         // global_load_b128
    *(float4*)&Bs[row * BSTRIDE + col] = v;               // ds_store_b128
  }
  __syncthreads();
}

// ---------------------------------------------------------------------------
// Kernel 1: Wh[h][n][o] = sum_f X[n][f] * W[h][f][o]
//   One wave handles a 16(M) x 64(O) strip of one head; K stepped by 4.
//   1024 wave-tasks = 8 heads * 128 strips. Grid 128 x block 256 (8 waves).
//   All 8 waves of a block share one head -> B tiles are block-shared in LDS.
// fp32 WMMA operand layout (ISA 7.12.2):
//   A 16x4 : lane<16 -> M=lane,    {K=k0,k0+1}; lane>=16 -> M=lane-16, {K=k0+2,k0+3}
//   B 4x16 : lane<16 -> N=lane,    {K=k0,k0+1}; lane>=16 -> N=lane-16, {K=k0+2,k0+3}
//   C 16x16: vgpr r  : lane<16 -> (M=r, N=lane); lane>=16 -> (M=r+8, N=lane-16)
// ---------------------------------------------------------------------------
__global__ void gat_gemm_xw(const float* __restrict__ X,   // [2048][512]
                            const float* __restrict__ W,   // [8][512][64]
                            float* __restrict__ Wh)        // [8][2048][64]
{
  __shared__ float Bs[KT * BSTRIDE];

  const int lane  = threadIdx.x & 31;
  const int wave  = (blockIdx.x * blockDim.x + threadIdx.x) >> 5;
  const int head  = wave >> 7;          // uniform per block (8 waves/block)
  const int strip = wave & 127;
  const int m0    = strip * 16;
  const int mA    = m0 + (lane & 15);
  const int kp    = (lane >> 4) * 2;    // 0 or 2
  const int nb    = lane & 15;

  const float* Bg = W + head * (NFEAT * NHID);
  const float* Xr = X + (size_t)mA * NFEAT;

  v8f c0 = {}, c1 = {}, c2 = {}, c3 = {};

  for (int kt = 0; kt < NFEAT; kt += KT) {
    stage_b_tile(Bg + (size_t)kt * NHID, Bs);
    __builtin_prefetch(Xr + kt + KT, 0, 3);     // next A chunk (speculative)

#pragma unroll 4
    for (int kk = 0; kk < KT; kk += 4) {
      v2f a = *(const v2f*)(Xr + kt + kk + kp); // two contiguous K values
      const float* Br = Bs + (kk + kp) * BSTRIDE + nb;
      v2f b0, b1, b2, b3;
      b0.x = Br[0];  b0.y = Br[BSTRIDE + 0];
      b1.x = Br[16]; b1.y = Br[BSTRIDE + 16];
      b2.x = Br[32]; b2.y = Br[BSTRIDE + 32];
      b3.x = Br[48]; b3.y = Br[BSTRIDE + 48];
      c0 = wmma_f32x4(a, b0, c0);
      c1 = wmma_f32x4(a, b1, c1);
      c2 = wmma_f32x4(a, b2, c2);
      c3 = wmma_f32x4(a, b3, c3);
    }
    __syncthreads();                            // before next tile overwrite
  }

  float* out = Wh + (size_t)head * N_NODES * NHID;
  const int mhi = 8 * (lane >> 4);
#pragma unroll
  for (int r = 0; r < 8; ++r) {
    const int m = m0 + r + mhi;
    float* orow = out + (size_t)m * NHID + nb;
    orow[0]  = c0[r];
    orow[16] = c1[r];
    orow[32] = c2[r];
    orow[48] = c3[r];
  }
}

// ---------------------------------------------------------------------------
// Kernel 2: s[h][n] = Wh[h][n][:] . a_src[h],  d[h][n] = Wh[h][n][:] . a_dst[h]
// ---------------------------------------------------------------------------
__global__ void gat_scores(const float* __restrict__ Wh,  // [8][2048][64]
                           const float* __restrict__ a,   // [8][128]
                           float* __restrict__ s,         // [8][2048]
                           float* __restrict__ d)         // [8][2048]
{
  const int idx = blockIdx.x * blockDim.x + threadIdx.x;  // 0..16383
  const int h = idx >> 11;
  const int n = idx & (N_NODES - 1);
  const float* wh   = Wh + ((size_t)h * N_NODES + n) * NHID;
  const float* asrc = a + h * 2 * NHID;
  const float* adst = asrc + NHID;
  float ss = 0.0f, dd = 0.0f;
#pragma unroll 8
  for (int o = 0; o < NHID; ++o) {
    const float w = wh[o];
    ss = fmaf(w, asrc[o], ss);
    dd = fmaf(w, adst[o], dd);
  }
  s[h * N_NODES + n] = ss;
  d[h * N_NODES + n] = dd;
}

// ---------------------------------------------------------------------------
// Kernel 3: masked leaky-relu softmax rows.  One block per row i; adj row is
// staged in LDS once and reused by all 8 heads. att -> d_out (it is returned).
// ---------------------------------------------------------------------------
__global__ void gat_softmax(const int* __restrict__ adj,  // [2048][2048]
                            const float* __restrict__ s,  // [8][2048]
                            const float* __restrict__ d,  // [8][2048]
                            float* __restrict__ att)      // [8][2048][2048]
{
  __shared__ int   adjrow[N_NODES];
  __shared__ float red[256];

  const int i = blockIdx.x;
  const int t = threadIdx.x;

  for (int j = t; j < N_NODES; j += 256)
    adjrow[j] = adj[(size_t)i * N_NODES + j];
  __syncthreads();

  for (int h = 0; h < NHEADS; ++h) {
    const float si = s[h * N_NODES + i];
    const float* dh = d + h * N_NODES;
    float ev[8];
    float lmax = -3.4e38f;
#pragma unroll
    for (int r = 0; r < 8; ++r) {
      const int j = t + r * 256;
      float e = si + dh[j];
      e = e > 0.0f ? e : LRELU_A * e;           // leaky relu
      e = adjrow[j] > 0 ? e : NEG_INF;          // adjacency mask
      ev[r] = e;
      lmax = fmaxf(lmax, e);
    }
    red[t] = lmax;
    __syncthreads();
    for (int off = 128; off > 0; off >>= 1) {
      if (t < off) red[t] = fmaxf(red[t], red[t + off]);
      __syncthreads();
    }
    const float m = red[0];
    __syncthreads();

    float lsum = 0.0f;
#pragma unroll
    for (int r = 0; r < 8; ++r) {
      ev[r] = expf(ev[r] - m);
      lsum += ev[r];
    }
    red[t] = lsum;
    __syncthreads();
    for (int off = 128; off > 0; off >>= 1) {
      if (t < off) red[t] += red[t + off];
      __syncthreads();
    }
    const float inv = 1.0f / red[0];
    __syncthreads();

    float* arow = att + ((size_t)h * N_NODES + i) * N_NODES;
#pragma unroll
    for (int r = 0; r < 8; ++r)
      arow[t + r * 256] = ev[r] * inv;          // coalesced
  }
}

// ---------------------------------------------------------------------------
// Kernel 4: Out[n][h*64+o] = elu(elu( sum_m att[h][n][m] * Wh[h][m][o] ))
//   Same WMMA strip structure as kernel 1, K = 2048, A row stride 2048.
//   B (Wh[head]) staged in LDS tiles; writes concat-heads [N, H*O] layout.
// ---------------------------------------------------------------------------
__global__ void gat_gemm_att(const float* __restrict__ Att, // [8][2048][2048]
                             const float* __restrict__ Wh,  // [8][2048][64]
                             float* __restrict__ Out)       // [2048][512]
{
  __shared__ float Bs[KT * BSTRIDE];

  const int lane  = threadIdx.x & 31;
  const int wave  = (blockIdx.x * blockDim.x + threadIdx.x) >> 5;
  const int head  = wave >> 7;          // uniform per block
  const int strip = wave & 127;
  const int m0    = strip * 16;
  const int mA    = m0 + (lane & 15);
  const int kp    = (lane >> 4) * 2;
  const int nb    = lane & 15;

  const float* Ar = Att + (size_t)head * N_NODES * N_NODES + (size_t)mA * N_NODES;
  const float* Bg = Wh  + (size_t)head * N_NODES * NHID;

  v8f c0 = {}, c1 = {}, c2 = {}, c3 = {};

  for (int kt = 0; kt < N_NODES; kt += KT) {
    stage_b_tile(Bg + (size_t)kt * NHID, Bs);
    __builtin_prefetch(Ar + kt + KT, 0, 3);     // next A chunk (speculative)

#pragma unroll 4
    for (int kk = 0; kk < KT; kk += 4) {
      v2f a = *(const v2f*)(Ar + kt + kk + kp);
      const float* Br = Bs + (kk + kp) * BSTRIDE + nb;
      v2f b0, b1, b2, b3;
      b0.x = Br[0];  b0.y = Br[BSTRIDE + 0];
      b1.x = Br[16]; b1.y = Br[BSTRIDE + 16];
      b2.x = Br[32]; b2.y = Br[BSTRIDE + 32];
      b3.x = Br[48]; b3.y = Br[BSTRIDE + 48];
      c0 = wmma_f32x4(a, b0, c0);
      c1 = wmma_f32x4(a, b1, c1);
      c2 = wmma_f32x4(a, b2, c2);
      c3 = wmma_f32x4(a, b3, c3);
    }
    __syncthreads();
  }

  const int mhi = 8 * (lane >> 4);
#pragma unroll
  for (int r = 0; r < 8; ++r) {
    const int m = m0 + r + mhi;
    float* orow = Out + (size_t)m * (NHEADS * NHID) + head * NHID + nb;
    orow[0]  = elu2(c0[r]);
    orow[16] = elu2(c1[r]);
    orow[32] = elu2(c2[r]);
    orow[48] = elu2(c3[r]);
  }
}

// ---------------------------------------------------------------------------
// Launch: two GAT layers, all on `stream`, graph-capture safe.
// d_out layout (return order): h2 [2048*512] | att1 [8*2048*2048] | att2 [...]
// ws layout (floats): Wh [8*2048*64] | s [8*2048] | d [8*2048] | h1 [2048*512]
// ---------------------------------------------------------------------------
extern "C" void kernel_launch(void* const* d_in, const int* in_sizes, int n_in,
                              void* d_out, int out_size, void* d_ws, size_t ws_size,
                              hipStream_t stream) {
  (void)in_sizes; (void)n_in; (void)out_size; (void)ws_size;

  const float* x   = (const float*)d_in[0];
  const int*   adj = (const int*)  d_in[1];
  const float* W1  = (const float*)d_in[2];
  const float* a1  = (const float*)d_in[3];
  const float* W2  = (const float*)d_in[4];
  const float* a2  = (const float*)d_in[5];

  float* out  = (float*)d_out;
  float* h2   = out;
  float* att1 = out + (size_t)N_NODES * NHEADS * NHID;                // +1,048,576
  float* att2 = att1 + (size_t)NHEADS * N_NODES * N_NODES;            // +33,554,432

  float* ws   = (float*)d_ws;
  float* Wh   = ws;                                                   // 1,048,576
  float* sbuf = Wh + (size_t)NHEADS * N_NODES * NHID;                 // 16,384
  float* dbuf = sbuf + NHEADS * N_NODES;                              // 16,384
  float* h1   = dbuf + NHEADS * N_NODES;                              // 1,048,576

  const dim3 gGemm(128), bGemm(256);   // 1024 waves = 8 heads * 128 strips
  const dim3 gSc(64),   bSc(256);      // 16384 (h,n) dot products
  const dim3 gSm(N_NODES), bSm(256);   // one block per row

  // ---- layer 1 ----
  gat_gemm_xw <<<gGemm, bGemm, 0, stream>>>(x,  W1, Wh);
  gat_scores  <<<gSc,   bSc,   0, stream>>>(Wh, a1, sbuf, dbuf);
  gat_softmax <<<gSm,   bSm,   0, stream>>>(adj, sbuf, dbuf, att1);
  gat_gemm_att<<<gGemm, bGemm, 0, stream>>>(att1, Wh, h1);

  // ---- layer 2 ----
  gat_gemm_xw <<<gGemm, bGemm, 0, stream>>>(h1, W2, Wh);
  gat_scores  <<<gSc,   bSc,   0, stream>>>(Wh, a2, sbuf, dbuf);
  gat_softmax <<<gSm,   bSm,   0, stream>>>(adj, sbuf, dbuf, att2);
  gat_gemm_att<<<gGemm, bGemm, 0, stream>>>(att2, Wh, h2);
}